// BiencoderRanker_57337813401642
// MI455X (gfx1250) — compile-verified
//
#include <hip/hip_runtime.h>
#include <hip/hip_bf16.h>

typedef float v2f __attribute__((ext_vector_type(2)));
typedef float v8f __attribute__((ext_vector_type(8)));

constexpr int kB = 8, kS = 512, kH = 768, kG = 3;
constexpr int kM = 5075;                    // number of candidate spans per batch
constexpr int kMT = (kM + 15) / 16;         // 318 mention tiles per batch
constexpr int kHT = kH / 16;                // 48 H tiles
constexpr int kWaves = 8;                   // 256 threads / wave32
constexpr int kHTPerWave = kHT / kWaves;    // 6 H tiles per wave

// One wave computes a 16(mention) x 16(H) f32 tile of the span-mean via
// V_WMMA_F32_16X16X4_F32:  E = W x X,  W[m,k] = in-span(m,k) ? 1/len(m) : 0.
__global__ __launch_bounds__(256)
void span_mean_wmma_kernel(const float* __restrict__ bert,   // (B,S,H)
                           const int*   __restrict__ bounds, // (B,M,2)
                           float*       __restrict__ out_emb)// (B*M,H)
{
    const int blk  = blockIdx.x;
    const int b    = blk / kMT;
    const int mt   = blk % kMT;
    const int m0   = mt * 16;

    const int tid    = threadIdx.x;
    const int wave   = tid >> 5;
    const int lane   = tid & 31;
    const int lane15 = lane & 15;
    const int hiHalf = lane >> 4;   // 0 -> K pair {0,1}; 1 -> K pair {2,3}

    // A-matrix row (mention) owned by this lane (same row in both lane halves).
    const int mA = m0 + lane15;
    int   sA = 0, eA = -1;
    float inv = 0.0f;
    if (mA < kM) {
        sA  = bounds[(b * kM + mA) * 2 + 0];
        eA  = bounds[(b * kM + mA) * 2 + 1];
        inv = 1.0f / (float)(eA - sA + 1);
    }

    // Wave-uniform K window over this mention tile (spans are short: width <= 12).
    int kmin = kS - 1, kmax = 0;
    for (int i = 0; i < 16; ++i) {
        int mm = m0 + i;
        if (mm < kM) {
            int s = bounds[(b * kM + mm) * 2 + 0];
            int e = bounds[(b * kM + mm) * 2 + 1];
            kmin = min(kmin, s);
            kmax = max(kmax, e);
        }
    }
    kmin = __builtin_amdgcn_readfirstlane(kmin);
    kmax = __builtin_amdgcn_readfirstlane(kmax);

    const int h0 = wave * (kHTPerWave * 16) + lane15;   // this lane's H column base

    v8f zero = {};
    v8f acc[kHTPerWave];
    #pragma unroll
    for (int t = 0; t < kHTPerWave; ++t) acc[t] = zero;

    const float* bert_b = bert + (size_t)b * kS * kH;

    for (int k0 = kmin; k0 <= kmax; k0 += 4) {
        // A layout (32-bit 16x4): lanes 0-15 hold K=k0,k0+1; lanes 16-31 hold K=k0+2,k0+3.
        const int kx0 = k0 + hiHalf * 2;
        const int kx1 = kx0 + 1;
        v2f a;
        a.x = (kx0 >= sA && kx0 <= eA) ? inv : 0.0f;
        a.y = (kx1 >= sA && kx1 <= eA) ? inv : 0.0f;

        // Clamp row index for in-bounds loads; weights are zero outside the window.
        const int kc0 = min(kx0, kS - 1);
        const int kc1 = min(kx1, kS - 1);
        const float* row0 = bert_b + (size_t)kc0 * kH;
        const float* row1 = bert_b + (size_t)kc1 * kH;

        #pragma unroll
        for (int t = 0; t < kHTPerWave; ++t) {
            v2f bm;                             // B layout mirrors A's K ordering
            bm.x = row0[h0 + t * 16];
            bm.y = row1[h0 + t * 16];
            acc[t] = __builtin_amdgcn_wmma_f32_16x16x4_f32(
                false, a, false, bm, (short)0, acc[t], false, false);
        }
    }

    // D layout (16x16 f32, 8 VGPRs): row = m0 + r + 8*hiHalf, col = lane15.
    #pragma unroll
    for (int t = 0; t < kHTPerWave; ++t) {
        #pragma unroll
        for (int r = 0; r < 8; ++r) {
            int m = m0 + r + hiHalf * 8;
            if (m < kM) {
                out_emb[((size_t)(b * kM + m)) * kH + (size_t)(h0 + t * 16)] = acc[t][r];
            }
        }
    }
}

// Scores copy, bounds-as-float copy, gold-span matching (last match wins).
__global__ void tail_outputs_kernel(const float* __restrict__ scores,
                                    const int*   __restrict__ bounds,
                                    const int*   __restrict__ gmb,    // (B,G,2)
                                    const unsigned char* __restrict__ gmask,
                                    const int*   __restrict__ gids,   // (B,G)
                                    float* __restrict__ out,
                                    size_t O2, size_t O3, size_t O4)
{
    int idx = blockIdx.x * blockDim.x + threadIdx.x;
    if (idx >= kB * kM) return;
    int b = idx / kM;

    out[O2 + idx] = scores[idx];

    int s = bounds[idx * 2 + 0];
    int e = bounds[idx * 2 + 1];
    out[O3 + (size_t)idx * 2 + 0] = (float)s;
    out[O3 + (size_t)idx * 2 + 1] = (float)e;

    bool found = false;
    int  id    = -1;
    for (int j = 0; j < kG; ++j) {
        int gs = gmb[(b * kG + j) * 2 + 0];
        int ge = gmb[(b * kG + j) * 2 + 1] - 1;   // reference subtracts 1 from end
        if (gmask[b * kG + j] == 0) { gs = -1; ge = -1; }
        if (gs >= 0 && s == gs && e == ge) { found = true; id = gids[b * kG + j]; }
    }
    out[O4 + idx] = found ? (float)id : -1.0f;
}

extern "C" void kernel_launch(void* const* d_in, const int* in_sizes, int n_in,
                              void* d_out, int out_size, void* d_ws, size_t ws_size,
                              hipStream_t stream) {
    const float* bert   = (const float*)d_in[0];          // (B,S,H) f32
    const float* scores = (const float*)d_in[1];          // (B,M)   f32
    const int*   bounds = (const int*)d_in[2];            // (B,M,2) i32
    const int*   gmb    = (const int*)d_in[3];            // (B,G,2) i32
    const unsigned char* gmask = (const unsigned char*)d_in[4]; // (B,G) bool
    const int*   gids   = (const int*)d_in[5];            // (B,G)   i32
    float* out = (float*)d_out;

    const size_t O2 = (size_t)kB * kM * kH;               // mention_scores
    const size_t O3 = O2 + (size_t)kB * kM;               // mention_bounds
    const size_t O4 = O3 + (size_t)kB * kM * 2;           // gold_id_extend

    span_mean_wmma_kernel<<<dim3(kB * kMT), 256, 0, stream>>>(bert, bounds, out);

    const int n2 = kB * kM;
    tail_outputs_kernel<<<dim3((n2 + 255) / 256), 256, 0, stream>>>(
        scores, bounds, gmb, gmask, gids, out, O2, O3, O4);
}